// CenterFaceMask_19937238188431
// MI455X (gfx1250) — compile-verified
//
#include <hip/hip_runtime.h>
#include <math.h>

#define HWDIM 512
#define HWPIX (512 * 512)
#define MOUT 128
#define NPTS 10
#define ABLOCKS 64                 // stage-1 blocks per channel
#define ACHUNK (HWPIX / ABLOCKS)   // 4096 elements per stage-1 block

typedef __attribute__((ext_vector_type(2))) float v2f;
typedef __attribute__((ext_vector_type(8))) float v8f;

// ---------------------------------------------------------------------------
// Argmax stage 1: 10*64 workgroups. Block (ch, blk) reduces a contiguous
// 4096-element slice of heat channel ch using float4 (b128) loads.
// Writes (maxVal, flatIdx) partials to workspace. Within a thread the
// candidate indices strictly increase, so `>` keeps the first occurrence;
// across threads the LDS reduction tie-breaks on the smaller index.
// ---------------------------------------------------------------------------
__global__ __launch_bounds__(256) void heat_argmax_stage1(
    const float* __restrict__ feat, float* __restrict__ partVal,
    int* __restrict__ partIdx) {
  __shared__ float sVal[256];
  __shared__ int sIdx[256];

  const int ch = blockIdx.x / ABLOCKS;
  const int blk = blockIdx.x % ABLOCKS;
  const int t = threadIdx.x;
  const float* base = feat + (size_t)(1027 + ch) * HWPIX + (size_t)blk * ACHUNK;

  float best = -INFINITY;
  int bidx = 0;
#pragma unroll
  for (int k = 0; k < 4; ++k) {
    const int vecIdx = k * 256 + t;                       // 0..1023 float4's
    const float4 v4 = ((const float4*)base)[vecIdx];
    const int e0 = blk * ACHUNK + vecIdx * 4;             // flat element index
    if (v4.x > best) { best = v4.x; bidx = e0 + 0; }
    if (v4.y > best) { best = v4.y; bidx = e0 + 1; }
    if (v4.z > best) { best = v4.z; bidx = e0 + 2; }
    if (v4.w > best) { best = v4.w; bidx = e0 + 3; }
  }
  sVal[t] = best;
  sIdx[t] = bidx;
  __syncthreads();

  for (int s = 128; s > 0; s >>= 1) {
    if (t < s) {
      const float v = sVal[t + s];
      const int i2 = sIdx[t + s];
      if (v > sVal[t] || (v == sVal[t] && i2 < sIdx[t])) {
        sVal[t] = v;
        sIdx[t] = i2;
      }
    }
    __syncthreads();
  }
  if (t == 0) {
    partVal[blockIdx.x] = sVal[0];
    partIdx[blockIdx.x] = sIdx[0];
  }
}

// ---------------------------------------------------------------------------
// Argmax stage 2: 10 blocks x 64 threads reduce the 64 partials per channel.
// Stage-1 blocks cover index-ordered slices, so the smaller-index tie-break
// yields exact jnp.argmax (first occurrence) semantics.
// ---------------------------------------------------------------------------
__global__ __launch_bounds__(64) void heat_argmax_stage2(
    const float* __restrict__ partVal, const int* __restrict__ partIdx,
    int* __restrict__ pts) {
  __shared__ float sVal[64];
  __shared__ int sIdx[64];

  const int ch = blockIdx.x;
  const int t = threadIdx.x;
  sVal[t] = partVal[ch * ABLOCKS + t];
  sIdx[t] = partIdx[ch * ABLOCKS + t];
  __syncthreads();

  for (int s = 32; s > 0; s >>= 1) {
    if (t < s) {
      const float v = sVal[t + s];
      const int i2 = sIdx[t + s];
      if (v > sVal[t] || (v == sVal[t] && i2 < sIdx[t])) {
        sVal[t] = v;
        sIdx[t] = i2;
      }
    }
    __syncthreads();
  }
  if (t == 0) {
    const int idx = sIdx[0];
    pts[ch * 2 + 0] = idx / HWDIM;  // py
    pts[ch * 2 + 1] = idx % HWDIM;  // px
  }
}

// ---------------------------------------------------------------------------
// Kernel 2: per-point bilinear resample as two f32 WMMA GEMMs + fused epilogue.
//   T (128x32)      = Wy (128x32) @ vec (32x32)
//   local (128x128) = T  (128x32) @ Wx^T (32x128)
//   out = valid ? sigmoid(local) * saliency[window] : 0
// One workgroup (256 threads = 8 wave32) per point.
// ---------------------------------------------------------------------------
__global__ __launch_bounds__(256) void bilinear_wmma_kernel(
    const float* __restrict__ feat, const int* __restrict__ pts,
    float* __restrict__ out) {
  __shared__ float sVec[32][33];   // 32x32 shape vector (padded)
  __shared__ float sWy[128][33];   // row-interp weights
  __shared__ float sWx[128][33];   // col-interp weights
  __shared__ float sT[128][33];    // intermediate Wy @ vec

  const int pt = blockIdx.x;  // 0..9
  const int t = threadIdx.x;
  const int py = pts[pt * 2 + 0];
  const int px = pts[pt * 2 + 1];
  const size_t pix = (size_t)py * HWDIM + px;

  // size = clip(int(|feat[1025/1026]|), 1, 128)
  int h = (int)fabsf(feat[(size_t)1025 * HWPIX + pix]);
  int w = (int)fabsf(feat[(size_t)1026 * HWPIX + pix]);
  h = min(max(h, 1), MOUT);
  w = min(max(w, 1), MOUT);

  // Gather the 1024 shape channels at (py,px) into a 32x32 tile.
  for (int c = t; c < 1024; c += 256) {
    sVec[c >> 5][c & 31] = feat[(size_t)(1 + c) * HWPIX + pix];
  }
  // Zero the dense weight matrices.
  for (int i = t; i < 128 * 32; i += 256) {
    sWy[i >> 5][i & 31] = 0.0f;
    sWx[i >> 5][i & 31] = 0.0f;
  }
  __syncthreads();

  // Fill bilinear taps: threads 0..127 -> Wy rows, 128..255 -> Wx rows.
  // If i0 == i1 (clamped edge) the += folds to a single weight of 1.0,
  // matching vec[y0]*(1-wy) + vec[y1]*wy with y0==y1.
  if (t < 128) {
    float s = fminf(fmaxf((t + 0.5f) * 32.0f / (float)h - 0.5f, 0.0f), 31.0f);
    int i0 = (int)floorf(s);
    int i1 = min(i0 + 1, 31);
    float f = s - (float)i0;
    sWy[t][i0] += 1.0f - f;
    sWy[t][i1] += f;
  } else {
    int r = t - 128;
    float s = fminf(fmaxf((r + 0.5f) * 32.0f / (float)w - 0.5f, 0.0f), 31.0f);
    int i0 = (int)floorf(s);
    int i1 = min(i0 + 1, 31);
    float f = s - (float)i0;
    sWx[r][i0] += 1.0f - f;
    sWx[r][i1] += f;
  }
  __syncthreads();

  const int wv = t >> 5;        // wave id 0..7
  const int lane = t & 31;
  const int m16 = lane & 15;    // row (A) / col (B,D) within 16
  const int g = lane >> 4;      // half-wave select

  // ---- GEMM1: T = Wy @ vec  (M=128, N=32, K=32) -> 16 tiles, 2 per wave.
  for (int tile = wv; tile < 16; tile += 8) {
    const int mt = tile >> 1;
    const int nt = tile & 1;
    v8f acc = {};
#pragma unroll
    for (int kk = 0; kk < 8; ++kk) {
      const int kb = kk * 4 + g * 2;
      v2f a, b;
      a.x = sWy[mt * 16 + m16][kb + 0];
      a.y = sWy[mt * 16 + m16][kb + 1];
      b.x = sVec[kb + 0][nt * 16 + m16];
      b.y = sVec[kb + 1][nt * 16 + m16];
      acc = __builtin_amdgcn_wmma_f32_16x16x4_f32(
          false, a, false, b, (short)0, acc, false, false);
    }
#pragma unroll
    for (int v = 0; v < 8; ++v) {
      sT[mt * 16 + g * 8 + v][nt * 16 + m16] = acc[v];
    }
  }
  __syncthreads();

  // ---- GEMM2: local = T @ Wx^T (M=128, N=128, K=32) -> 64 tiles, 8 per wave.
  const float* __restrict__ sal = feat;  // channel 0
  float* __restrict__ optr = out + (size_t)pt * (MOUT * MOUT);
  const int r0 = py - h / 2;
  const int c0 = px - w / 2;

  for (int tile = wv; tile < 64; tile += 8) {
    const int mt = tile >> 3;
    const int nt = tile & 7;
    v8f acc = {};
#pragma unroll
    for (int kk = 0; kk < 8; ++kk) {
      const int kb = kk * 4 + g * 2;
      v2f a, b;
      a.x = sT[mt * 16 + m16][kb + 0];
      a.y = sT[mt * 16 + m16][kb + 1];
      // B[k][n] = Wx[nt*16 + n][k]
      b.x = sWx[nt * 16 + m16][kb + 0];
      b.y = sWx[nt * 16 + m16][kb + 1];
      acc = __builtin_amdgcn_wmma_f32_16x16x4_f32(
          false, a, false, b, (short)0, acc, false, false);
    }
#pragma unroll
    for (int v = 0; v < 8; ++v) {
      const int r = mt * 16 + g * 8 + v;
      const int s = nt * 16 + m16;
      const float local = 1.0f / (1.0f + expf(-acc[v]));
      const int gr = r0 + r;
      const int gc = c0 + s;
      const bool valid =
          (r < h) && (s < w) && (gr >= 0) && (gr < HWDIM) && (gc >= 0) && (gc < HWDIM);
      float o = 0.0f;
      if (valid) o = local * sal[(size_t)gr * HWDIM + gc];
      optr[r * MOUT + s] = o;
    }
  }
}

extern "C" void kernel_launch(void* const* d_in, const int* in_sizes, int n_in,
                              void* d_out, int out_size, void* d_ws, size_t ws_size,
                              hipStream_t stream) {
  const float* feat = (const float*)d_in[0];
  float* out = (float*)d_out;

  // Workspace layout: 640 floats partVal | 640 ints partIdx | 20 ints pts
  float* partVal = (float*)d_ws;
  int* partIdx = (int*)(partVal + NPTS * ABLOCKS);
  int* pts = partIdx + NPTS * ABLOCKS;

  heat_argmax_stage1<<<NPTS * ABLOCKS, 256, 0, stream>>>(feat, partVal, partIdx);
  heat_argmax_stage2<<<NPTS, 64, 0, stream>>>(partVal, partIdx, pts);
  bilinear_wmma_kernel<<<NPTS, 256, 0, stream>>>(feat, pts, out);
}